// HybridQLSTM_65481071396724
// MI455X (gfx1250) — compile-verified
//
#include <hip/hip_runtime.h>
#include <math.h>
#include <stdint.h>

// ---------------------------------------------------------------------------
// HybridQLSTM for MI455X (gfx1250, wave32)
//   K1 quanv: pair-factorized 2-qubit sims (exact)      -> feats (2048,784)
//   K2 pack:  Bt[n][k] = W_{n/4}[n%4][k]  (16 x 784)
//   K3 gemm:  xpre = feats @ Bt^T  via V_WMMA_F32_16X16X4_F32,
//             B panel staged to LDS with TENSOR_LOAD_TO_LDS (TDM)
//   K4 recur: 128-step scan, ONE wave32, closed-form qlayer
//   K5 tag:   logits + log_softmax -> d_out (128,16,10)
// ---------------------------------------------------------------------------

typedef float    v2f  __attribute__((ext_vector_type(2)));
typedef float    v8f  __attribute__((ext_vector_type(8)));
typedef uint32_t u32x4 __attribute__((ext_vector_type(4)));
typedef uint32_t u32x8 __attribute__((ext_vector_type(8)));

// ---------------- 2-qubit pair-state helpers (4 complex amps) ---------------
// index = hi*2 + lo.  HI gate acts on pairs (0,2),(1,3); LO on (0,1),(2,3).

template <bool HI>
__device__ __forceinline__ void ry4(float (&re)[4], float (&im)[4], float t) {
  float c = cosf(0.5f * t), s = sinf(0.5f * t);
  const int i0 = 0, j0 = HI ? 2 : 1, i1 = HI ? 1 : 2, j1 = 3;
  float ar, ai, br, bi;
  ar = re[i0]; ai = im[i0]; br = re[j0]; bi = im[j0];
  re[i0] = c * ar - s * br; im[i0] = c * ai - s * bi;
  re[j0] = s * ar + c * br; im[j0] = s * ai + c * bi;
  ar = re[i1]; ai = im[i1]; br = re[j1]; bi = im[j1];
  re[i1] = c * ar - s * br; im[i1] = c * ai - s * bi;
  re[j1] = s * ar + c * br; im[j1] = s * ai + c * bi;
}

template <bool HI>
__device__ __forceinline__ void rx4(float (&re)[4], float (&im)[4], float t) {
  float c = cosf(0.5f * t), s = sinf(0.5f * t);
  const int i0 = 0, j0 = HI ? 2 : 1, i1 = HI ? 1 : 2, j1 = 3;
  float ar, ai, br, bi;
  ar = re[i0]; ai = im[i0]; br = re[j0]; bi = im[j0];
  re[i0] = c * ar + s * bi; im[i0] = c * ai - s * br;
  re[j0] = s * ai + c * br; im[j0] = c * bi - s * ar;
  ar = re[i1]; ai = im[i1]; br = re[j1]; bi = im[j1];
  re[i1] = c * ar + s * bi; im[i1] = c * ai - s * br;
  re[j1] = s * ai + c * br; im[j1] = c * bi - s * ar;
}

template <bool HI>
__device__ __forceinline__ void rz4(float (&re)[4], float (&im)[4], float t) {
  float c = cosf(0.5f * t), s = sinf(0.5f * t);
#pragma unroll
  for (int i = 0; i < 4; ++i) {
    int bit = HI ? (i >> 1) : (i & 1);
    float r = re[i], m = im[i];
    if (bit == 0) { re[i] = r * c + m * s; im[i] = m * c - r * s; }
    else          { re[i] = r * c - m * s; im[i] = m * c + r * s; }
  }
}

__device__ __forceinline__ void cnot4(float (&re)[4], float (&im)[4]) {
  // control = hi bit, target = lo bit: swap amps 2 <-> 3
  float tr = re[2], ti = im[2];
  re[2] = re[3]; im[2] = im[3];
  re[3] = tr;    im[3] = ti;
}

// ---------------- K1: quanv (pair-factorized, exact) ------------------------
// Circuit couples (w0,w1) and (w2,w3) independently; final CNOT(1,2) only
// XORs labels: out = [ZA_hi, ZA_lo, ZA_lo*ZB_hi, ZB_lo].
__global__ void k_quanv(const float* __restrict__ img,
                        const float* __restrict__ rl,
                        float* __restrict__ feats) {
  int id = blockIdx.x * blockDim.x + threadIdx.x;
  if (id >= 2048 * 196) return;
  int n = id / 196, p = id - n * 196;
  int r = p / 14, c = p - r * 14;
  const float* base = img + n * 784 + (r * 2) * 28 + c * 2;
  float p0 = base[0], p1 = base[1], p2 = base[28], p3 = base[29];
  float r0 = rl[0], r1 = rl[1], r2 = rl[2], r3 = rl[3], r4 = rl[4];

  // pair A: wires (0,1): RY(p0)hi RY(p1)lo RX(r0)hi RY(r1)lo CNOT RY(r4)hi
  float are[4] = {1.f, 0.f, 0.f, 0.f}, aim[4] = {0.f, 0.f, 0.f, 0.f};
  ry4<true >(are, aim, p0);
  ry4<false>(are, aim, p1);
  rx4<true >(are, aim, r0);
  ry4<false>(are, aim, r1);
  cnot4(are, aim);
  ry4<true >(are, aim, r4);
  float pa0 = are[0]*are[0]+aim[0]*aim[0], pa1 = are[1]*are[1]+aim[1]*aim[1];
  float pa2 = are[2]*are[2]+aim[2]*aim[2], pa3 = are[3]*are[3]+aim[3]*aim[3];
  float zahi = pa0 + pa1 - pa2 - pa3;  // Z on wire0
  float zalo = pa0 - pa1 + pa2 - pa3;  // Z on wire1

  // pair B: wires (2,3): RY(p2)hi RY(p3)lo RZ(r2)hi RX(r3)lo CNOT
  float bre[4] = {1.f, 0.f, 0.f, 0.f}, bim[4] = {0.f, 0.f, 0.f, 0.f};
  ry4<true >(bre, bim, p2);
  ry4<false>(bre, bim, p3);
  rz4<true >(bre, bim, r2);
  rx4<false>(bre, bim, r3);
  cnot4(bre, bim);
  float pb0 = bre[0]*bre[0]+bim[0]*bim[0], pb1 = bre[1]*bre[1]+bim[1]*bim[1];
  float pb2 = bre[2]*bre[2]+bim[2]*bim[2], pb3 = bre[3]*bre[3]+bim[3]*bim[3];
  float zbhi = pb0 + pb1 - pb2 - pb3;  // Z on wire2
  float zblo = pb0 - pb1 + pb2 - pb3;  // Z on wire3

  float* o = feats + n * 784 + p * 4;
  o[0] = zahi;
  o[1] = zalo;
  o[2] = zalo * zbhi;   // wire2 after CNOT(1,2)
  o[3] = zblo;
}

// ---------------- K2: pack stacked gate x-weights, transposed ----------------
__global__ void k_pack(const float* __restrict__ Wf, const float* __restrict__ Wi,
                       const float* __restrict__ Wu, const float* __restrict__ Wo,
                       float* __restrict__ Bt) {
  int id = blockIdx.x * blockDim.x + threadIdx.x;
  if (id >= 16 * 784) return;
  int n = id / 784, k = id - n * 784;
  const float* W = (n < 4) ? Wf : (n < 8) ? Wi : (n < 12) ? Wu : Wo;
  Bt[n * 784 + k] = W[(n & 3) * 788 + k];
}

// ---------------- K3: GEMM (2048x784)x(784x16), f32 WMMA + TDM ---------------
// B panel (16x784 f32, 50KB) DMA'd once per block into LDS by the Tensor
// Data Mover; A streams from global.  Fragment layout per ISA 7.12.2.
__global__ void __launch_bounds__(256) k_gemm(const float* __restrict__ feats,
                                              const float* __restrict__ Bt,
                                              float* __restrict__ xpre) {
  __shared__ float lds_b[16 * 784];

  const int lane = threadIdx.x & 31;
  const int wave = threadIdx.x >> 5;
  const int tile = blockIdx.x * (blockDim.x >> 5) + wave; // 0..127
  const int mn = lane & 15;
  const int khalf = (lane >> 4) << 1; // 0 or 2

  // LDS byte offset of lds_b.  The ptrtoint both fills the descriptor's
  // lds_addr field and ESCAPES the LDS object into the asm below, so the
  // compiler must assume the TDM writes it (otherwise the array is DCE'd).
  const uint32_t lds_off = (uint32_t)(size_t)&lds_b[0];

  if (wave == 0) {
    // Tensor DMA descriptor (ISA ch.8): 2D tensor 784x16 f32, whole-tile load.
    uint64_t ga = (uint64_t)(uintptr_t)Bt;
    u32x4 g0;
    g0[0] = 1u;                                   // count=1 (valid), user mode
    g0[1] = lds_off;                              // lds_addr (bytes)
    g0[2] = (uint32_t)ga;                         // global_addr[31:0]
    g0[3] = ((uint32_t)(ga >> 32) & 0x1FFFFFFu)   // global_addr[56:32]
            | 0x80000000u;                        // type=2 ("image")
    u32x8 g1;
    g1[0] = 2u << 16;                // data_size=2 (4 bytes)
    g1[1] = (784u & 0xFFFFu) << 16;  // tensor_dim0[15:0] at bits 63:48
    g1[2] = 16u << 16;               // tensor_dim0 hi=0 | tensor_dim1 lo16=16
    g1[3] = 784u << 16;              // tensor_dim1 hi=0 | tile_dim0=784
    g1[4] = 16u;                     // tile_dim1=16, tile_dim2=0
    g1[5] = 784u;                    // tensor_dim0_stride lo32=784
    g1[6] = 0u;                      // stride hi | tensor_dim1_stride lo
    g1[7] = 0u;
    asm volatile("tensor_load_to_lds %1, %2"
                 :
                 : "v"(lds_off), "s"(g0), "s"(g1)
                 : "memory");
    __builtin_amdgcn_s_wait_tensorcnt(0);
  }
  __syncthreads();

  const float* arow = feats + (tile * 16 + mn) * 784;
  const float* brow = lds_b + mn * 784;
  __builtin_prefetch(arow, 0, 0);

  v8f acc = {};
#pragma unroll 4
  for (int k = 0; k < 784; k += 4) {
    v2f a = *(const v2f*)(arow + k + khalf);
    v2f b = *(const v2f*)(brow + k + khalf);
    acc = __builtin_amdgcn_wmma_f32_16x16x4_f32(
        /*neg_a=*/false, a, /*neg_b=*/false, b,
        /*c_mod=*/(short)0, acc, /*reuse_a=*/false, /*reuse_b=*/false);
  }

  const int rowoff = (lane >> 4) * 8;
  const int col = lane & 15;
#pragma unroll
  for (int r = 0; r < 8; ++r) {
    xpre[(tile * 16 + rowoff + r) * 16 + col] = acc[r];
  }
}

// ---------------- K4: sequential QLSTM scan, ONE wave32 ----------------------
// Closed form: qlayer(ang,q) with z_w = cos(ang_w + q_w) gives
//   [z1*z2*z3, z0*z1, z0*z1*z2, z0*z1*z2*z3].
// Lane t: gate sims (g, c) for g in {t>>4, (t>>4)+2}, c = t&15;
//         cell-state units (t>>2, t&3) and (t>>2 + 8, t&3).
__global__ void __launch_bounds__(32) k_recur(
    const float* __restrict__ xpre,
    const float* __restrict__ Wf, const float* __restrict__ Wi,
    const float* __restrict__ Wu, const float* __restrict__ Wo,
    const float* __restrict__ bf, const float* __restrict__ bi,
    const float* __restrict__ bu, const float* __restrict__ bo,
    const float* __restrict__ qf, const float* __restrict__ qi,
    const float* __restrict__ qu, const float* __restrict__ qo,
    float* __restrict__ outs) {
  __shared__ float hx[16][4];
  __shared__ float gv[4][16][4];

  const int t = threadIdx.x;        // 0..31 (one wave)
  const int c = t & 15;
  const int gbase = t >> 4;         // this lane's gates: gbase, gbase+2

  float wh[2][4][4], bias[2][4], qp[2][4];
#pragma unroll
  for (int s = 0; s < 2; ++s) {
    const int G = gbase + 2 * s;
    const float* W  = (G == 0) ? Wf : (G == 1) ? Wi : (G == 2) ? Wu : Wo;
    const float* bb = (G == 0) ? bf : (G == 1) ? bi : (G == 2) ? bu : bo;
    const float* qq = (G == 0) ? qf : (G == 1) ? qi : (G == 2) ? qu : qo;
#pragma unroll
    for (int h = 0; h < 4; ++h) {
      bias[s][h] = bb[h];
      qp[s][h] = qq[h];
#pragma unroll
      for (int j = 0; j < 4; ++j) wh[s][h][j] = W[h * 788 + 784 + j];
    }
  }

  const int c1 = t >> 2, h1 = t & 3;
  const int c2 = c1 + 8;
  float cx1 = 0.f, cx2 = 0.f;
  hx[c1][h1] = 0.f;
  hx[c2][h1] = 0.f;
  __syncthreads();

  for (int step = 0; step < 128; ++step) {
    float hx0 = hx[c][0], hxv1 = hx[c][1], hxv2 = hx[c][2], hxv3 = hx[c][3];
#pragma unroll
    for (int s = 0; s < 2; ++s) {
      const int G = gbase + 2 * s;
      const float4 xp = *(const float4*)(xpre + (step * 16 + c) * 16 + G * 4);
      const float xph[4] = {xp.x, xp.y, xp.z, xp.w};
      float z[4];
#pragma unroll
      for (int h = 0; h < 4; ++h) {
        float a = xph[h] + bias[s][h] + hx0 * wh[s][h][0] + hxv1 * wh[s][h][1]
                + hxv2 * wh[s][h][2] + hxv3 * wh[s][h][3];
        z[h] = cosf(a + qp[s][h]);
      }
      float v1 = z[0] * z[1];
      float v2 = v1 * z[2];
      float v3 = v2 * z[3];
      float v0 = z[1] * z[2] * z[3];
      float vv[4] = {v0, v1, v2, v3};
#pragma unroll
      for (int h = 0; h < 4; ++h) {
        float v = vv[h];
        v = (G == 2) ? tanhf(v) : (1.f / (1.f + expf(-v)));
        gv[G][c][h] = v;
      }
    }
    __syncthreads();

    // cell updates (two units per lane)
    {
      float fv = gv[0][c1][h1], iv = gv[1][c1][h1];
      float uv = gv[2][c1][h1], ov = gv[3][c1][h1];
      cx1 = fv * cx1 + iv * uv;
      float hv = ov * tanhf(cx1);
      hx[c1][h1] = hv;
      outs[step * 64 + c1 * 4 + h1] = hv;
    }
    {
      float fv = gv[0][c2][h1], iv = gv[1][c2][h1];
      float uv = gv[2][c2][h1], ov = gv[3][c2][h1];
      cx2 = fv * cx2 + iv * uv;
      float hv = ov * tanhf(cx2);
      hx[c2][h1] = hv;
      outs[step * 64 + c2 * 4 + h1] = hv;
    }
    __syncthreads();
  }
}

// ---------------- K5: tag linear + log_softmax -------------------------------
__global__ void k_tag(const float* __restrict__ outs,
                      const float* __restrict__ Wtag,
                      const float* __restrict__ btag,
                      float* __restrict__ out) {
  int row = blockIdx.x * blockDim.x + threadIdx.x; // 0..2047 = b*16 + s
  if (row >= 2048) return;
  float h0 = outs[row * 4 + 0], h1 = outs[row * 4 + 1];
  float h2 = outs[row * 4 + 2], h3 = outs[row * 4 + 3];
  float l[10];
  float m = -1e30f;
#pragma unroll
  for (int j = 0; j < 10; ++j) {
    float v = btag[j] + h0 * Wtag[j * 4 + 0] + h1 * Wtag[j * 4 + 1] +
              h2 * Wtag[j * 4 + 2] + h3 * Wtag[j * 4 + 3];
    l[j] = v;
    m = fmaxf(m, v);
  }
  float se = 0.f;
#pragma unroll
  for (int j = 0; j < 10; ++j) se += expf(l[j] - m);
  float lse = m + logf(se);
#pragma unroll
  for (int j = 0; j < 10; ++j) out[row * 10 + j] = l[j] - lse;
}

// ---------------------------------------------------------------------------
extern "C" void kernel_launch(void* const* d_in, const int* in_sizes, int n_in,
                              void* d_out, int out_size, void* d_ws, size_t ws_size,
                              hipStream_t stream) {
  const float* images = (const float*)d_in[0];
  const float* rl     = (const float*)d_in[1];
  const float* qf     = (const float*)d_in[2];
  const float* qi     = (const float*)d_in[3];
  const float* qu     = (const float*)d_in[4];
  const float* qo     = (const float*)d_in[5];
  const float* Wf     = (const float*)d_in[6];
  const float* bf     = (const float*)d_in[7];
  const float* Wi     = (const float*)d_in[8];
  const float* bi     = (const float*)d_in[9];
  const float* Wu     = (const float*)d_in[10];
  const float* bu     = (const float*)d_in[11];
  const float* Wo     = (const float*)d_in[12];
  const float* bo     = (const float*)d_in[13];
  const float* Wtag   = (const float*)d_in[14];
  const float* btag   = (const float*)d_in[15];
  float* out = (float*)d_out;

  char* ws = (char*)d_ws;
  float* feats = (float*)(ws);                                            // 2048*784
  float* Bt    = (float*)(ws + 2048 * 784 * 4);                           // 16*784
  float* xpre  = (float*)(ws + 2048 * 784 * 4 + 16 * 784 * 4);            // 2048*16
  float* outs  = (float*)(ws + 2048 * 784 * 4 + 16 * 784 * 4 + 2048 * 16 * 4); // 128*64

  k_quanv<<<(2048 * 196) / 256, 256, 0, stream>>>(images, rl, feats);
  k_pack<<<(16 * 784 + 255) / 256, 256, 0, stream>>>(Wf, Wi, Wu, Wo, Bt);
  k_gemm<<<16, 256, 0, stream>>>(feats, Bt, xpre);
  k_recur<<<1, 32, 0, stream>>>(xpre, Wf, Wi, Wu, Wo, bf, bi, bu, bo,
                                qf, qi, qu, qo, outs);
  k_tag<<<8, 256, 0, stream>>>(outs, Wtag, btag, out);
}